// EDModel_60584808677653
// MI455X (gfx1250) — compile-verified
//
#include <hip/hip_runtime.h>
#include <hip/hip_bf16.h>

typedef __attribute__((ext_vector_type(2))) float v2f;
typedef __attribute__((ext_vector_type(8))) float v8f;

#define BB 16
#define SS 128
#define TT (BB * SS)          // 2048 tokens
#define WED 300
#define BIOD 25
#define POSD 25
#define DEPD 50
#define RH 256
#define GD 300
#define NLAY 2
#define NTAGS 34
#define IN_DIM (WED + BIOD)   // 325
#define G4 (4 * RH)           // 1024
#define HWROW (2 * GD + DEPD) // 650

static __device__ __forceinline__ float sigmoidf_(float x) {
    return 1.0f / (1.0f + __expf(-x));
}

// ---------------------------------------------------------------------------
// Generic tiled f32 GEMM on WMMA 16x16x4.
//   C[m,n] = act( sum_k A[m,k] * B(k,n) + bias1[n] + bias2[n] )
//   transB=1: B(k,n) = Bm[n*ldb + k]  (weight stored as (N,K), i.e. A @ W^T)
//   transB=0: B(k,n) = Bm[k*ldb + n]
//   batch via blockIdx.z with strides. act: 0=none, 1=ELU.
//
// CONTRACT: M must be a multiple of BM (32). All call sites obey this
// (2048, 128, 262144), which lets us drop every M-bound check.
//
// Block: 256 threads (8 waves). Tile: 32(M) x 64(N); wave (wr,wc) does 16x16.
// K is split into full 32-wide stages plus one zero-padded checked tail.
// Staging is two-phase: batched unconditional (clamped) global loads into
// registers, then batched LDS stores -> loads stay in flight together.
// 8 v_wmma per barrier pair.
// ---------------------------------------------------------------------------
#define BM 32
#define BN 64
#define BKK 32
#define A_SL ((BM * BKK) / 256)   // 4 slices/thread
#define B_SL ((BN * BKK) / 256)   // 8 slices/thread

__global__ __launch_bounds__(256) void gemm_f32_wmma(
    const float* __restrict__ A, int lda, long long strideA,
    const float* __restrict__ Bm, int ldb, long long strideB, int transB,
    const float* __restrict__ bias1, const float* __restrict__ bias2,
    float* __restrict__ C, int ldc, long long strideC,
    int M, int N, int K, int act)
{
    __shared__ float As[BM][BKK + 1];
    __shared__ float Bs[BN][BKK + 1];   // Bs[n][k]

    const int bz = blockIdx.z;
    A  += (long long)bz * strideA;
    Bm += (long long)bz * strideB;
    C  += (long long)bz * strideC;

    const int m0   = blockIdx.y * BM;
    const int n0   = blockIdx.x * BN;
    const int tid  = threadIdx.x;
    const int lane = tid & 31;
    const int wave = tid >> 5;
    const int wr   = wave >> 2;   // 0..1 : M sub-tile
    const int wc   = wave & 3;    // 0..3 : N sub-tile

    v8f acc = {0.0f, 0.0f, 0.0f, 0.0f, 0.0f, 0.0f, 0.0f, 0.0f};

    const int row   = lane & 15;          // M row (A) / N col (B) within tile
    const int khalf = (lane >> 4) << 1;   // 0 for lanes 0-15, 2 for 16-31

    const int Kfull = K & ~(BKK - 1);

    int k0 = 0;
    for (; k0 < Kfull; k0 += BKK) {
        float ra[A_SL], rb[B_SL];
        // phase 1: gather to registers (unconditional; A fully in-bounds,
        // B address clamped to row N-1 so always valid)
#pragma unroll
        for (int s = 0; s < A_SL; ++s) {
            int i = tid + s * 256;
            int r = i >> 5, c = i & 31;
            ra[s] = A[(long long)(m0 + r) * lda + (k0 + c)];
        }
#pragma unroll
        for (int s = 0; s < B_SL; ++s) {
            int i = tid + s * 256;
            int r = i >> 5, c = i & 31;
            int gn = n0 + r;
            int cn = gn < N ? gn : N - 1;
            rb[s] = transB ? Bm[(long long)cn * ldb + (k0 + c)]
                           : Bm[(long long)(k0 + c) * ldb + cn];
        }
        // prefetch next K tile while this stage's loads complete
        if (k0 + BKK < K) {
            __builtin_prefetch(A + (long long)(m0 + (tid & 31)) * lda + (k0 + BKK), 0, 3);
            int pn = n0 + (tid & 63); pn = pn < N ? pn : N - 1;
            const float* bp = transB ? (Bm + (long long)pn * ldb + (k0 + BKK))
                                     : (Bm + (long long)(k0 + BKK) * ldb + pn);
            __builtin_prefetch(bp, 0, 3);
        }
        // phase 2: batched LDS stores (N-select applied here, not on loads)
#pragma unroll
        for (int s = 0; s < A_SL; ++s) {
            int i = tid + s * 256;
            As[i >> 5][i & 31] = ra[s];
        }
#pragma unroll
        for (int s = 0; s < B_SL; ++s) {
            int i = tid + s * 256;
            int r = i >> 5;
            Bs[r][i & 31] = (n0 + r < N) ? rb[s] : 0.0f;
        }
        __syncthreads();
#pragma unroll
        for (int kk = 0; kk < BKK; kk += 4) {
            v2f a, b;
            a.x = As[wr * 16 + row][kk + khalf];
            a.y = As[wr * 16 + row][kk + khalf + 1];
            b.x = Bs[wc * 16 + row][kk + khalf];
            b.y = Bs[wc * 16 + row][kk + khalf + 1];
            acc = __builtin_amdgcn_wmma_f32_16x16x4_f32(
                false, a, false, b, (short)0, acc, false, false);
        }
        __syncthreads();
    }

    // ---- checked tail stage (K % 32 != 0) ----
    if (k0 < K) {
        float ra[A_SL], rb[B_SL];
#pragma unroll
        for (int s = 0; s < A_SL; ++s) {
            int i = tid + s * 256;
            int r = i >> 5, c = i & 31;
            int gk = k0 + c;
            int ck = gk < K ? gk : K - 1;
            ra[s] = A[(long long)(m0 + r) * lda + ck];
        }
#pragma unroll
        for (int s = 0; s < B_SL; ++s) {
            int i = tid + s * 256;
            int r = i >> 5, c = i & 31;
            int gn = n0 + r, gk = k0 + c;
            int cn = gn < N ? gn : N - 1;
            int ck = gk < K ? gk : K - 1;
            rb[s] = transB ? Bm[(long long)cn * ldb + ck]
                           : Bm[(long long)ck * ldb + cn];
        }
#pragma unroll
        for (int s = 0; s < A_SL; ++s) {
            int i = tid + s * 256;
            int c = i & 31;
            As[i >> 5][c] = (k0 + c < K) ? ra[s] : 0.0f;
        }
#pragma unroll
        for (int s = 0; s < B_SL; ++s) {
            int i = tid + s * 256;
            int r = i >> 5, c = i & 31;
            Bs[r][c] = (n0 + r < N && k0 + c < K) ? rb[s] : 0.0f;
        }
        __syncthreads();
#pragma unroll
        for (int kk = 0; kk < BKK; kk += 4) {
            v2f a, b;
            a.x = As[wr * 16 + row][kk + khalf];
            a.y = As[wr * 16 + row][kk + khalf + 1];
            b.x = Bs[wc * 16 + row][kk + khalf];
            b.y = Bs[wc * 16 + row][kk + khalf + 1];
            acc = __builtin_amdgcn_wmma_f32_16x16x4_f32(
                false, a, false, b, (short)0, acc, false, false);
        }
    }

    // C/D layout: VGPR v: lanes 0-15 -> (M=v, N=lane); lanes 16-31 -> (M=v+8).
    // M is in-bounds by contract; only N needs a check.
    const int nl   = lane & 15;
    const int mofs = (lane >> 4) * 8;
    const int gn   = n0 + wc * 16 + nl;
    if (gn < N) {
        float bv = (bias1 ? bias1[gn] : 0.0f) + (bias2 ? bias2[gn] : 0.0f);
#pragma unroll
        for (int v = 0; v < 8; ++v) {
            int gm = m0 + wr * 16 + mofs + v;
            float x = acc[v] + bv;
            if (act == 1) x = (x > 0.0f) ? x : expm1f(x);
            C[(long long)gm * ldc + gn] = x;
        }
    }
}

// ---------------------------------------------------------------------------
// Embedding gathers
// ---------------------------------------------------------------------------
__global__ void embed_x_kernel(const int* __restrict__ wseq,
                               const int* __restrict__ ets,
                               const float* __restrict__ wtab,
                               const float* __restrict__ btab,
                               float* __restrict__ x_emb)
{
    int t = blockIdx.x * blockDim.x + threadIdx.x;
    if (t >= TT * IN_DIM) return;
    int tok = t / IN_DIM, c = t % IN_DIM;
    float v = (c < WED) ? wtab[(long long)wseq[tok] * WED + c]
                        : btab[(long long)ets[tok] * BIOD + (c - WED)];
    x_emb[t] = v;
}

__global__ void embed_pos_kernel(const int* __restrict__ pos,
                                 const float* __restrict__ ptab,
                                 float* __restrict__ pos_e)
{
    int t = blockIdx.x * blockDim.x + threadIdx.x;
    if (t >= TT * POSD) return;
    int tok = t / POSD, c = t % POSD;
    pos_e[t] = ptab[(long long)pos[tok] * POSD + c];
}

__global__ void embed_wadj_kernel(const int* __restrict__ dep,
                                  const float* __restrict__ etab,
                                  float* __restrict__ wadj)
{
    long long t = (long long)blockIdx.x * blockDim.x + threadIdx.x;
    if (t >= (long long)BB * SS * SS * DEPD) return;
    long long bij = t / DEPD;
    int e = (int)(t % DEPD);
    wadj[t] = etab[(long long)dep[bij] * DEPD + e];
}

// ---------------------------------------------------------------------------
// Bidirectional LSTM recurrence. One workgroup per (batch, direction).
// 256 threads == RH; thread t owns hidden unit t. h shared via LDS.
// ---------------------------------------------------------------------------
__global__ __launch_bounds__(RH) void lstm_kernel(
    const float* __restrict__ xw_f, const float* __restrict__ xw_b,
    const float* __restrict__ Whh_f, const float* __restrict__ Whh_b,
    float* __restrict__ rnn_out)
{
    const int b   = blockIdx.x;
    const int dir = blockIdx.y;
    const float* xw  = dir ? xw_b  : xw_f;
    const float* Whh = dir ? Whh_b : Whh_f;

    __shared__ float hsh[RH];
    const int t = threadIdx.x;
    hsh[t] = 0.0f;
    float c = 0.0f;

    const float* wi = Whh + (long long)(0 * RH + t) * RH;
    const float* wf = Whh + (long long)(1 * RH + t) * RH;
    const float* wg = Whh + (long long)(2 * RH + t) * RH;
    const float* wo = Whh + (long long)(3 * RH + t) * RH;

    for (int s = 0; s < SS; ++s) {
        const int st = dir ? (SS - 1 - s) : s;
        __syncthreads();
        const float* xr = xw + (long long)(b * SS + st) * G4;
        float gi = xr[t], gf = xr[RH + t], gg = xr[2 * RH + t], go = xr[3 * RH + t];
        for (int j = 0; j < RH; ++j) {
            float hj = hsh[j];
            gi = fmaf(wi[j], hj, gi);
            gf = fmaf(wf[j], hj, gf);
            gg = fmaf(wg[j], hj, gg);
            go = fmaf(wo[j], hj, go);
        }
        c = sigmoidf_(gf) * c + sigmoidf_(gi) * tanhf(gg);
        float hn = sigmoidf_(go) * tanhf(c);
        __syncthreads();
        hsh[t] = hn;
        rnn_out[(long long)(b * SS + st) * (2 * RH) + dir * RH + t] = hn;
    }
}

// ---------------------------------------------------------------------------
// Asum[b,i,j] = (1/DEPD) * sum_e wadj[b,i,j,e]
// Block-cooperative: stage 256 rows (256*50 floats) coalesced into LDS, then
// each thread reduces its own row out of LDS.
// ---------------------------------------------------------------------------
#define ASUM_ROWS 256
__global__ __launch_bounds__(256) void asum_kernel(const float* __restrict__ wadj,
                                                   float* __restrict__ Asum)
{
    __shared__ float buf[ASUM_ROWS * DEPD];   // 51.2 KB of the 320 KB LDS
    const long long base = (long long)blockIdx.x * (ASUM_ROWS * DEPD);
    const int tid = threadIdx.x;
#pragma unroll
    for (int i = tid; i < ASUM_ROWS * DEPD; i += 256)
        buf[i] = wadj[base + i];
    __syncthreads();
    const float* p = buf + tid * DEPD;
    float s = 0.0f;
#pragma unroll
    for (int e = 0; e < DEPD; ++e) s += p[e];
    Asum[(long long)blockIdx.x * ASUM_ROWS + tid] = s * (1.0f / (float)DEPD);
}

// ---------------------------------------------------------------------------
// wadj[b,i,j,f] += N1[(b,j),f] + N2[(b,i),f] + hb[f]
// ---------------------------------------------------------------------------
__global__ void hw_add_kernel(float* __restrict__ wadj,
                              const float* __restrict__ N1,
                              const float* __restrict__ N2,
                              const float* __restrict__ hb)
{
    long long t = (long long)blockIdx.x * blockDim.x + threadIdx.x;
    if (t >= (long long)BB * SS * SS * DEPD) return;
    int f = (int)(t % DEPD);
    long long bij = t / DEPD;
    int j = (int)(bij % SS);
    long long bi = bij / SS;
    int i = (int)(bi % SS);
    int b = (int)(bi / SS);
    wadj[t] += N1[(long long)(b * SS + j) * DEPD + f] +
               N2[(long long)(b * SS + i) * DEPD + f] + hb[f];
}

// ---------------------------------------------------------------------------
// outputs[tok, d] = (pos_e[tok,:] . gW[d,:] + gb[d]) * node[tok, d]
// (fused tiny K=25 gate GEMM + elementwise multiply; ld_out = 3*GD)
// ---------------------------------------------------------------------------
__global__ void gate_mul_kernel(const float* __restrict__ pos_e,
                                const float* __restrict__ gW,
                                const float* __restrict__ gb,
                                const float* __restrict__ node,
                                float* __restrict__ outp)
{
    int t = blockIdx.x * blockDim.x + threadIdx.x;
    if (t >= TT * GD) return;
    int tok = t / GD, d = t % GD;
    const float* pe = pos_e + (long long)tok * POSD;
    const float* wr = gW + (long long)d * POSD;
    float g = gb[d];
    for (int p = 0; p < POSD; ++p) g = fmaf(pe[p], wr[p], g);
    outp[(long long)tok * ((NLAY + 1) * GD) + d] = g * node[(long long)tok * GD + d];
}

// ---------------------------------------------------------------------------
// Host orchestration
// ---------------------------------------------------------------------------
static inline void launch_gemm(hipStream_t stream,
                               const float* A, int lda, long long sA,
                               const float* Bm, int ldb, long long sB, int transB,
                               const float* b1, const float* b2,
                               float* C, int ldc, long long sC,
                               int M, int N, int K, int act, int batch)
{
    dim3 grid((N + BN - 1) / BN, (M + BM - 1) / BM, batch);
    hipLaunchKernelGGL(gemm_f32_wmma, grid, dim3(256), 0, stream,
                       A, lda, sA, Bm, ldb, sB, transB, b1, b2,
                       C, ldc, sC, M, N, K, act);
}

extern "C" void kernel_launch(void* const* d_in, const int* in_sizes, int n_in,
                              void* d_out, int out_size, void* d_ws, size_t ws_size,
                              hipStream_t stream)
{
    const int* word_seq = (const int*)d_in[0];
    const int* ets      = (const int*)d_in[2];
    const int* dep      = (const int*)d_in[4];
    const int* pos      = (const int*)d_in[5];
    const float* word_table = (const float*)d_in[6];
    const float* bio_table  = (const float*)d_in[7];
    const float* pos_table  = (const float*)d_in[8];
    const float* edge_table = (const float*)d_in[9];
    const float* Wih_f = (const float*)d_in[10];
    const float* Whh_f = (const float*)d_in[11];
    const float* bih_f = (const float*)d_in[12];
    const float* bhh_f = (const float*)d_in[13];
    const float* Wih_b = (const float*)d_in[14];
    const float* Whh_b = (const float*)d_in[15];
    const float* bih_b = (const float*)d_in[16];
    const float* bhh_b = (const float*)d_in[17];
    const float* Wg    = (const float*)d_in[18];
    const float* bg    = (const float*)d_in[19];
    const float* gate_W = (const float*)d_in[20];
    const float* gate_b = (const float*)d_in[21];
    const float* gcn_W  = (const float*)d_in[22];
    const float* gcn_b  = (const float*)d_in[23];
    const float* hw_W   = (const float*)d_in[24];
    const float* hw_b   = (const float*)d_in[25];
    const float* agg_W  = (const float*)d_in[26];
    const float* agg_b  = (const float*)d_in[27];
    const float* cls_W  = (const float*)d_in[28];
    const float* cls_b  = (const float*)d_in[29];

    float* ws = (float*)d_ws;
    size_t o = 0;
    float* x_emb   = ws + o; o += (size_t)TT * IN_DIM;
    float* pos_e   = ws + o; o += (size_t)TT * POSD;
    float* xw_f    = ws + o; o += (size_t)TT * G4;
    float* xw_b    = ws + o; o += (size_t)TT * G4;
    float* rnn_out = ws + o; o += (size_t)TT * 2 * RH;
    float* gcn_in  = ws + o; o += (size_t)TT * GD;
    float* wadj    = ws + o; o += (size_t)BB * SS * SS * DEPD;
    float* Asum    = ws + o; o += (size_t)BB * SS * SS;
    float* ax      = ws + o; o += (size_t)TT * GD;
    float* node    = ws + o; o += (size_t)TT * GD;
    float* N1      = ws + o; o += (size_t)TT * DEPD;
    float* N2      = ws + o; o += (size_t)TT * DEPD;
    float* outputs = ws + o; o += (size_t)TT * (NLAY + 1) * GD;
    float* agg     = ws + o; o += (size_t)TT * GD;

    float* logits_out = (float*)d_out;                        // 2048 x 34
    float* wadj_out   = (float*)d_out + (size_t)TT * NTAGS;   // final wadj

    // 1. Embeddings
    hipLaunchKernelGGL(embed_x_kernel, dim3((TT * IN_DIM + 255) / 256), dim3(256),
                       0, stream, word_seq, ets, word_table, bio_table, x_emb);
    hipLaunchKernelGGL(embed_pos_kernel, dim3((TT * POSD + 255) / 256), dim3(256),
                       0, stream, pos, pos_table, pos_e);
    {
        long long tot = (long long)BB * SS * SS * DEPD;
        hipLaunchKernelGGL(embed_wadj_kernel, dim3((unsigned)((tot + 255) / 256)),
                           dim3(256), 0, stream, dep, edge_table, wadj);
    }

    // 2. LSTM input GEMMs: xw = x_emb @ Wih^T + bih + bhh
    launch_gemm(stream, x_emb, IN_DIM, 0, Wih_f, IN_DIM, 0, 1, bih_f, bhh_f,
                xw_f, G4, 0, TT, G4, IN_DIM, 0, 1);
    launch_gemm(stream, x_emb, IN_DIM, 0, Wih_b, IN_DIM, 0, 1, bih_b, bhh_b,
                xw_b, G4, 0, TT, G4, IN_DIM, 0, 1);

    // 3. LSTM recurrence (fwd + bwd), writes concatenated rnn_out
    hipLaunchKernelGGL(lstm_kernel, dim3(BB, 2), dim3(RH), 0, stream,
                       xw_f, xw_b, Whh_f, Whh_b, rnn_out);

    // 4. gcn_in = rnn_out @ Wg^T + bg
    launch_gemm(stream, rnn_out, 2 * RH, 0, Wg, 2 * RH, 0, 1, bg, nullptr,
                gcn_in, GD, 0, TT, GD, 2 * RH, 0, 1);

    // 5. layer_list[0] = gate0 * gcn_in
    hipLaunchKernelGGL(gate_mul_kernel, dim3((TT * GD + 255) / 256), dim3(256),
                       0, stream, pos_e, gate_W, gate_b, gcn_in, outputs);

    const float* cur_node = gcn_in;
    for (int l = 0; l < NLAY; ++l) {
        // a. A[b,i,j] = sum_e wadj / DEPD
        hipLaunchKernelGGL(asum_kernel, dim3(BB * SS * SS / ASUM_ROWS), dim3(256),
                           0, stream, wadj, Asum);
        // b. ax[b] = Asum[b] @ node[b]   (batched)
        launch_gemm(stream, Asum, SS, (long long)SS * SS,
                    cur_node, GD, (long long)SS * GD, 0, nullptr, nullptr,
                    ax, GD, (long long)SS * GD, SS, GD, SS, 0, BB);
        // c. node = elu(ax @ gcn_W[l]^T + gcn_b[l])
        launch_gemm(stream, ax, GD, 0, gcn_W + (size_t)l * GD * GD, GD, 0, 1,
                    gcn_b + (size_t)l * GD, nullptr, node, GD, 0, TT, GD, GD, 1, 1);
        // d. N1 = node @ Wn1^T ; N2 = node @ Wn2^T (slices of hw_W[l])
        const float* hwl = hw_W + (size_t)l * DEPD * HWROW;
        launch_gemm(stream, node, GD, 0, hwl + DEPD, HWROW, 0, 1, nullptr, nullptr,
                    N1, DEPD, 0, TT, DEPD, GD, 0, 1);
        launch_gemm(stream, node, GD, 0, hwl + DEPD + GD, HWROW, 0, 1, nullptr, nullptr,
                    N2, DEPD, 0, TT, DEPD, GD, 0, 1);
        // e. wadj' = wadj @ We^T  (in-place for l=0; writes d_out for l=1)
        float* wdst = (l == NLAY - 1) ? wadj_out : wadj;
        launch_gemm(stream, wadj, DEPD, 0, hwl, HWROW, 0, 1, nullptr, nullptr,
                    wdst, DEPD, 0, BB * SS * SS, DEPD, DEPD, 0, 1);
        // f. += n1[j] + n2[i] + hw_b[l]
        {
            long long tot = (long long)BB * SS * SS * DEPD;
            hipLaunchKernelGGL(hw_add_kernel, dim3((unsigned)((tot + 255) / 256)),
                               dim3(256), 0, stream, wdst, N1, N2,
                               hw_b + (size_t)l * DEPD);
        }
        // g. layer_list[l+1] = gate_{l+1} * node
        hipLaunchKernelGGL(gate_mul_kernel, dim3((TT * GD + 255) / 256), dim3(256),
                           0, stream, pos_e, gate_W + (size_t)(l + 1) * GD * POSD,
                           gate_b + (size_t)(l + 1) * GD, node,
                           outputs + (size_t)(l + 1) * GD);
        cur_node = node;
    }

    // 7. agg = outputs @ agg_W^T + agg_b
    launch_gemm(stream, outputs, (NLAY + 1) * GD, 0, agg_W, (NLAY + 1) * GD, 0, 1,
                agg_b, nullptr, agg, GD, 0, TT, GD, (NLAY + 1) * GD, 0, 1);
    // 8. logits = agg @ cls_W^T + cls_b  -> front of d_out
    launch_gemm(stream, agg, GD, 0, cls_W, GD, 0, 1, cls_b, nullptr,
                logits_out, NTAGS, 0, TT, NTAGS, GD, 0, 1);
}